// DGI_893353197864
// MI455X (gfx1250) — compile-verified
//
#include <hip/hip_runtime.h>
#include <math.h>

typedef __attribute__((ext_vector_type(2))) float v2f;
typedef __attribute__((ext_vector_type(8))) float v8f;

#define FDIM 64

// ---------------- degree / normalization ----------------
__global__ void k_init_deg(float* __restrict__ deg, int N) {
    int i = blockIdx.x * blockDim.x + threadIdx.x;
    if (i < N) deg[i] = 1.0f;          // self-loop contributes 1
}

__global__ void k_zero64(float* __restrict__ p) {
    if (threadIdx.x < 64) p[threadIdx.x] = 0.0f;
}

__global__ void k_count_deg(const int* __restrict__ rows, float* __restrict__ deg, int E) {
    int e = blockIdx.x * blockDim.x + threadIdx.x;
    if (e < E) atomicAdd(&deg[rows[e]], 1.0f);
}

__global__ void k_rsqrt(float* __restrict__ deg, int N) {
    int i = blockIdx.x * blockDim.x + threadIdx.x;
    if (i < N) deg[i] = rsqrtf(deg[i]);   // deg >= 1 always (self loops)
}

// ---------------- WMMA GEMM: Y[M,64] = gather(X, perm) @ W[64,64] ----------------
// One wave computes a 16-row x 64-col output stripe via V_WMMA_F32_16X16X4_F32.
__global__ void k_gemm64_wmma(const float* __restrict__ X,
                              const int*  __restrict__ perm,   // may be null
                              const float* __restrict__ W,
                              float* __restrict__ Y, int M) {
    int gtid = blockIdx.x * blockDim.x + threadIdx.x;
    int wave = gtid >> 5;
    int lane = threadIdx.x & 31;
    int row0 = wave * 16;
    if (row0 >= M) return;               // wave-uniform exit (EXEC stays all-1s)

    int half = lane >> 4;                // 0: lanes 0-15, 1: lanes 16-31
    int l    = lane & 15;

    int r   = row0 + l;
    int src = perm ? perm[r] : r;
    const float* xrow = X + (size_t)src * FDIM;

    v8f acc0 = {}, acc1 = {}, acc2 = {}, acc3 = {};

    #pragma unroll
    for (int k = 0; k < 64; k += 4) {
        // A fragment 16x4: lane half selects K pair {k,k+1} vs {k+2,k+3}
        v2f a;
        a.x = xrow[k + 2 * half + 0];
        a.y = xrow[k + 2 * half + 1];

        const float* wr0 = W + (size_t)(k + 2 * half + 0) * FDIM;
        const float* wr1 = W + (size_t)(k + 2 * half + 1) * FDIM;

        v2f b0, b1, b2, b3;
        b0.x = wr0[l +  0]; b0.y = wr1[l +  0];
        b1.x = wr0[l + 16]; b1.y = wr1[l + 16];
        b2.x = wr0[l + 32]; b2.y = wr1[l + 32];
        b3.x = wr0[l + 48]; b3.y = wr1[l + 48];

        acc0 = __builtin_amdgcn_wmma_f32_16x16x4_f32(false, a, false, b0, (short)0, acc0, false, false);
        acc1 = __builtin_amdgcn_wmma_f32_16x16x4_f32(false, a, false, b1, (short)0, acc1, false, false);
        acc2 = __builtin_amdgcn_wmma_f32_16x16x4_f32(false, a, false, b2, (short)0, acc2, false, false);
        acc3 = __builtin_amdgcn_wmma_f32_16x16x4_f32(false, a, false, b3, (short)0, acc3, false, false);
    }

    // C/D layout: VGPR r -> row (row0 + r + 8*half), col (n0 + l)
    #pragma unroll
    for (int rr = 0; rr < 8; ++rr) {
        size_t out_row = (size_t)(row0 + rr + 8 * half) * FDIM;
        Y[out_row + l +  0] = acc0[rr];
        Y[out_row + l + 16] = acc1[rr];
        Y[out_row + l + 32] = acc2[rr];
        Y[out_row + l + 48] = acc3[rr];
    }
}

// ---------------- aggregation ----------------
// Self-loop term initializes the accumulator: agg[i,:] = xw[i,:] * dis[i]^2
__global__ void k_selfloop_init(const float* __restrict__ xw,
                                const float* __restrict__ dis,
                                float* __restrict__ agg, int N) {
    int t = blockIdx.x * blockDim.x + threadIdx.x;   // N*16 threads, float4 each
    int i  = t >> 4;
    int f4 = (t & 15) << 2;
    if (i >= N) return;
    float w = dis[i] * dis[i];
    const float4 v = *(const float4*)(xw + (size_t)i * FDIM + f4);
    float4 o;
    o.x = v.x * w; o.y = v.y * w; o.z = v.z * w; o.w = v.w * w;
    *(float4*)(agg + (size_t)i * FDIM + f4) = o;
}

// Edge scatter: agg[r,:] += xw[c,:] * dis[r]*dis[c]  (16 threads / edge, float4)
__global__ void k_edge_scatter(const float* __restrict__ xw,
                               const int*  __restrict__ rows,
                               const int*  __restrict__ cols,
                               const float* __restrict__ dis,
                               float* __restrict__ agg, int E) {
    int t = blockIdx.x * blockDim.x + threadIdx.x;
    int e  = t >> 4;
    int f4 = (t & 15) << 2;
    if (e >= E) return;
    int r = rows[e];
    int c = cols[e];
    float norm = dis[r] * dis[c];
    const float4 v = *(const float4*)(xw + (size_t)c * FDIM + f4);
    float* dst = agg + (size_t)r * FDIM + f4;
    atomicAdd(dst + 0, v.x * norm);
    atomicAdd(dst + 1, v.y * norm);
    atomicAdd(dst + 2, v.z * norm);
    atomicAdd(dst + 3, v.w * norm);
}

// h = prelu(agg + b)  (in place)
__global__ void k_bias_prelu(float* __restrict__ h, const float* __restrict__ b,
                             const float* __restrict__ a_ptr, int N) {
    int t = blockIdx.x * blockDim.x + threadIdx.x;
    int i  = t >> 4;
    int f4 = (t & 15) << 2;
    if (i >= N) return;
    float a = *a_ptr;
    float4 v  = *(float4*)(h + (size_t)i * FDIM + f4);
    const float4 bb = *(const float4*)(b + f4);
    v.x += bb.x; v.y += bb.y; v.z += bb.z; v.w += bb.w;
    v.x = (v.x >= 0.0f) ? v.x : a * v.x;
    v.y = (v.y >= 0.0f) ? v.y : a * v.y;
    v.z = (v.z >= 0.0f) ? v.z : a * v.z;
    v.w = (v.w >= 0.0f) ? v.w : a * v.w;
    *(float4*)(h + (size_t)i * FDIM + f4) = v;
}

// out += b  (in place, no activation)
__global__ void k_bias(float* __restrict__ h, const float* __restrict__ b, int N) {
    int t = blockIdx.x * blockDim.x + threadIdx.x;
    int i  = t >> 4;
    int f4 = (t & 15) << 2;
    if (i >= N) return;
    float4 v  = *(float4*)(h + (size_t)i * FDIM + f4);
    const float4 bb = *(const float4*)(b + f4);
    v.x += bb.x; v.y += bb.y; v.z += bb.z; v.w += bb.w;
    *(float4*)(h + (size_t)i * FDIM + f4) = v;
}

// ---------------- summary ----------------
__global__ void k_summary_partial(const float* __restrict__ pos,
                                  float* __restrict__ ssum, int N) {
    int t = blockIdx.x * blockDim.x + threadIdx.x;
    int f      = t & 63;
    int stripe = t >> 6;
    int nstripes = (gridDim.x * blockDim.x) >> 6;
    float s = 0.0f;
    for (int i = stripe; i < N; i += nstripes)
        s += pos[(size_t)i * FDIM + f];
    atomicAdd(&ssum[f], s);
}

__global__ void k_summary_final(const float* __restrict__ ssum,
                                float* __restrict__ out, float invN) {
    int f = threadIdx.x;
    if (f < 64) {
        float m = ssum[f] * invN;
        out[f] = 1.0f / (1.0f + expf(-m));
    }
}

// ---------------- orchestration ----------------
extern "C" void kernel_launch(void* const* d_in, const int* in_sizes, int n_in,
                              void* d_out, int out_size, void* d_ws, size_t ws_size,
                              hipStream_t stream) {
    const float* x    = (const float*)d_in[0];
    const int*   ei   = (const int*)  d_in[1];
    const int*   perm = (const int*)  d_in[2];
    const float* W1   = (const float*)d_in[3];
    const float* b1   = (const float*)d_in[4];
    const float* pa   = (const float*)d_in[5];   // prelu_a (1 element)
    const float* W2   = (const float*)d_in[6];
    const float* b2   = (const float*)d_in[7];

    const int N = in_sizes[0] / FDIM;
    const int E = in_sizes[1] / 2;
    const int* rows = ei;
    const int* cols = ei + E;

    float* out = (float*)d_out;
    float* out_pos = out;
    float* out_neg = out + (size_t)N * FDIM;
    float* out_sum = out + (size_t)2 * N * FDIM;

    // Workspace: dis[N] | ssum[64] | bufA[N*64] | bufB[N*64]
    float* dis  = (float*)d_ws;
    float* ssum = dis + N;
    float* bufA = ssum + 64;
    float* bufB = bufA + (size_t)N * FDIM;

    const int TPB = 256;
    int blkN    = (N + TPB - 1) / TPB;
    int blkE    = (E + TPB - 1) / TPB;
    int blkN16  = (N * 16 + TPB - 1) / TPB;
    int blkE16  = ((long long)E * 16 + TPB - 1) / TPB;
    int waves   = (N + 15) / 16;                 // one wave per 16 output rows
    int blkGemm = (waves * 32 + TPB - 1) / TPB;

    // degree -> dis
    k_init_deg<<<blkN, TPB, 0, stream>>>(dis, N);
    k_zero64<<<1, 64, 0, stream>>>(ssum);
    k_count_deg<<<blkE, TPB, 0, stream>>>(rows, dis, E);
    k_rsqrt<<<blkN, TPB, 0, stream>>>(dis, N);

    for (int branch = 0; branch < 2; ++branch) {
        const int* p = (branch == 0) ? (const int*)nullptr : perm;
        float* ob    = (branch == 0) ? out_pos : out_neg;

        // layer 1: xw = gather(x,p) @ W1 -> bufA ; aggregate -> bufB ; prelu
        k_gemm64_wmma<<<blkGemm, TPB, 0, stream>>>(x, p, W1, bufA, N);
        k_selfloop_init<<<blkN16, TPB, 0, stream>>>(bufA, dis, bufB, N);
        k_edge_scatter<<<blkE16, TPB, 0, stream>>>(bufA, rows, cols, dis, bufB, E);
        k_bias_prelu<<<blkN16, TPB, 0, stream>>>(bufB, b1, pa, N);

        // layer 2: xw2 = bufB @ W2 -> bufA ; aggregate -> out ; + b2
        k_gemm64_wmma<<<blkGemm, TPB, 0, stream>>>(bufB, nullptr, W2, bufA, N);
        k_selfloop_init<<<blkN16, TPB, 0, stream>>>(bufA, dis, ob, N);
        k_edge_scatter<<<blkE16, TPB, 0, stream>>>(bufA, rows, cols, dis, ob, E);
        k_bias<<<blkN16, TPB, 0, stream>>>(ob, b2, N);
    }

    // summary = sigmoid(mean(positive, axis=0))
    k_summary_partial<<<64, TPB, 0, stream>>>(out_pos, ssum, N);
    k_summary_final<<<1, 64, 0, stream>>>(ssum, out_sum, 1.0f / (float)N);
}